// PCDONN_88450556494247
// MI455X (gfx1250) — compile-verified
//
#include <hip/hip_runtime.h>

// ============================================================================
// PCDONN on MI455X (gfx1250): diffractive ONN forward pass.
// DFT-as-GEMM formulation so the 4 propagations run on the WMMA pipe:
//   FFT2(f)  = W f W          (W = 512x512 symmetric DFT matrix, fp32)
//   IFFT2(g) = Wb g Wb        (Wb = conj(W)/512)
// Complex GEMM with the 3-multiplication (Gauss) trick:
//   Prr += Ar*Br; Pii += Ai*Bi; Px += (Ar+Ai)*(Br+Bi)
//   Cr = Prr - Pii ; Ci = Px - Prr - Pii
// Each wave owns a 16x32 output tile -> 6 f32 WMMA accumulators; one A fetch
// feeds 6 v_wmma_f32_16x16x4_f32 per k-step. The LDS B panel is stored
// K-pair-interleaved ([kpair][n][2]) so every WMMA B fragment is a single
// aligned ds_load_b64 into an even VGPR pair (no shuffle moves).
// d_ws layout (floats): Wf_r|Wf_i|Wb_r|Wb_i|H_r|H_i  (6 * 256K)
//                       U_r|U_i|V_r|V_i               (4 * 144 * 256K)
// Total ~611 MB workspace.
// ============================================================================

typedef __attribute__((ext_vector_type(2))) float v2f;
typedef __attribute__((ext_vector_type(8))) float v8f;

#define DIM   512
#define NPIX  (DIM * DIM)
#define NMODE 9
#define NBAT  16
#define NIMG  (NMODE * NBAT)
#define NCLS  10
#define PI_D  3.14159265358979323846

// ---------------- constants: DFT matrices + angular-spectrum transfer -------
__global__ __launch_bounds__(256) void init_consts_kernel(
    float* __restrict__ wfr, float* __restrict__ wfi,
    float* __restrict__ wbr, float* __restrict__ wbi,
    float* __restrict__ hr,  float* __restrict__ hi) {
  int idx = blockIdx.x * blockDim.x + threadIdx.x;
  if (idx >= NPIX) return;
  int i = idx >> 9, j = idx & (DIM - 1);
  // Forward DFT: W[i][j] = exp(-2*pi*1j * i*j / 512); reduce i*j mod 512 first.
  int t = (i * j) & (DIM - 1);
  double ang = -2.0 * PI_D * (double)t / (double)DIM;
  double ca = cos(ang), sa = sin(ang);
  wfr[idx] = (float)ca;
  wfi[idx] = (float)sa;
  wbr[idx] = (float)(ca / (double)DIM);   // conj(W)/512
  wbi[idx] = (float)(-sa / (double)DIM);
  // Angular-spectrum transfer H (same for all 4 props: z == 0.05 everywhere)
  const double lam = 5.32e-7, d = 8e-6, z = 0.05;
  double fx = (double)((i < 256) ? i : i - 512) / ((double)DIM * d);
  double fy = (double)((j < 256) ? j : j - 512) / ((double)DIM * d);
  double arg = 1.0 - (lam * fx) * (lam * fx) - (lam * fy) * (lam * fy);
  double sq = (arg > 0.0) ? sqrt(arg) : 0.0;
  double ph = (2.0 * PI_D / lam) * z * sq;
  hr[idx] = (float)cos(ph);
  hi[idx] = (float)sin(ph);
}

// ---------------- initial field: x_amp * exp(i * noise * (m+1)/2) -----------
__global__ __launch_bounds__(256) void init_field_kernel(
    const float* __restrict__ x_amp, const float* __restrict__ pnoise,
    float* __restrict__ ur, float* __restrict__ ui) {
  size_t idx = (size_t)blockIdx.x * blockDim.x + threadIdx.x;
  if (idx >= (size_t)NIMG * NPIX) return;
  int img = (int)(idx / NPIX);
  int p   = (int)(idx % NPIX);
  int m   = img / NBAT;
  int b   = img % NBAT;
  float fac = 0.5f * (float)(m + 1);
  float s, c;
  sincosf(pnoise[idx] * fac, &s, &c);
  float a = x_amp[(size_t)b * NPIX + p];
  ur[idx] = a * c;
  ui[idx] = a * s;
}

// ---------------- elementwise: field *= amplitude * exp(i*phase) ------------
__global__ __launch_bounds__(256) void ew_layer_kernel(
    float* __restrict__ ur, float* __restrict__ ui,
    const float* __restrict__ amp, const float* __restrict__ phs) {
  size_t idx = (size_t)blockIdx.x * blockDim.x + threadIdx.x;
  if (idx >= (size_t)NIMG * NPIX) return;
  int p = (int)(idx % NPIX);
  float s, c;
  sincosf(phs[p], &s, &c);
  float a  = amp[p];
  float tr = a * c, ti = a * s;
  float r = ur[idx], im = ui[idx];
  ur[idx] = r * tr - im * ti;
  ui[idx] = r * ti + im * tr;
}

// ---------------- elementwise: field *= H (precomputed complex) -------------
__global__ __launch_bounds__(256) void ew_h_kernel(
    float* __restrict__ ur, float* __restrict__ ui,
    const float* __restrict__ hr, const float* __restrict__ hi) {
  size_t idx = (size_t)blockIdx.x * blockDim.x + threadIdx.x;
  if (idx >= (size_t)NIMG * NPIX) return;
  int p = (int)(idx % NPIX);
  float tr = hr[p], ti = hi[p];
  float r = ur[idx], im = ui[idx];
  ur[idx] = r * tr - im * ti;
  ui[idx] = r * ti + im * tr;
}

// ---------------- complex GEMM via f32 WMMA (C = A x B, per image) ----------
// Block = (32,8): 8 wave32s, each owning one 16(M)x32(N) output tile.
// strideA/strideB: per-image element stride (0 => operand shared, e.g. W).
// B K-panel (64x32, re+im) staged in 16 KB LDS, K-pair interleaved:
//   element (k, n) lives at flat index (( (k>>1)*32 + n )*2 + (k&1)).
__global__ __launch_bounds__(256) void gemm_cplx_kernel(
    const float* __restrict__ ar_g, const float* __restrict__ ai_g,
    unsigned long long strideA,
    const float* __restrict__ br_g, const float* __restrict__ bi_g,
    unsigned long long strideB,
    float* __restrict__ cr_g, float* __restrict__ ci_g) {
  __shared__ float lbr[64 * 32];
  __shared__ float lbi[64 * 32];

  const int lane = threadIdx.x;              // 0..31 (wave32)
  const int wave = threadIdx.y;              // 0..7
  const int tid  = wave * 32 + lane;
  const int n0   = blockIdx.x * 32;
  const int m0   = (blockIdx.y * 8 + wave) * 16;
  const int img  = blockIdx.z;

  const float* __restrict__ Ar = ar_g + (size_t)img * strideA;
  const float* __restrict__ Ai = ai_g + (size_t)img * strideA;
  const float* __restrict__ Br = br_g + (size_t)img * strideB;
  const float* __restrict__ Bi = bi_g + (size_t)img * strideB;

  // f32 16x4 A layout: lanes 0-15 hold M=lane, K=(0,1); lanes 16-31 K=(2,3)
  const int arow = m0 + (lane & 15);
  const int koff = (lane >> 4) * 2;
  // B 4x16 layout mirrors A's K split; C/D: lane&15 = N, (lane>>4)*8+v = M
  const int bn   = lane & 15;

  // Gauss 3-mult accumulators, two 16x16 N-subtiles each
  v8f rr0 = {}, ii0 = {}, xx0 = {};
  v8f rr1 = {}, ii1 = {}, xx1 = {};

  for (int kc = 0; kc < DIM; kc += 64) {
    // cooperative LDS fill (K-pair interleaved): thread e owns (kpair, n),
    // loads B[kc+2kp][n0+n] and B[kc+2kp+1][n0+n], stores one 8B pair.
    for (int e = tid; e < 32 * 32; e += 256) {
      int kp = e >> 5, n = e & 31;
      size_t g = (size_t)(kc + 2 * kp) * DIM + (n0 + n);
      v2f pr = { Br[g], Br[g + DIM] };
      v2f pi = { Bi[g], Bi[g + DIM] };
      *reinterpret_cast<v2f*>(&lbr[e * 2]) = pr;
      *reinterpret_cast<v2f*>(&lbi[e * 2]) = pi;
    }
    if (tid < 64 && kc + 64 < DIM) {   // prefetch next panel (global_prefetch_b8)
      __builtin_prefetch(Br + (size_t)(kc + 64 + tid) * DIM + n0, 0, 1);
      __builtin_prefetch(Bi + (size_t)(kc + 64 + tid) * DIM + n0, 0, 1);
    }
    __syncthreads();

#pragma unroll
    for (int kk = 0; kk < 64; kk += 4) {
      size_t ga = (size_t)arow * DIM + (kc + kk + koff);   // 8B-aligned
      v2f a_r = *reinterpret_cast<const v2f*>(Ar + ga);
      v2f a_i = *reinterpret_cast<const v2f*>(Ai + ga);
      v2f a_s = a_r + a_i;                                  // v_pk_add_f32
      // fragment base: kpair = (kk+koff)/2, 8B-aligned ds_load_b64 each
      int fb0 = (((kk + koff) >> 1) * 32 + bn) * 2;         // subtile 0
      v2f br0 = *reinterpret_cast<const v2f*>(&lbr[fb0]);
      v2f bi0 = *reinterpret_cast<const v2f*>(&lbi[fb0]);
      v2f bs0 = br0 + bi0;
      v2f br1 = *reinterpret_cast<const v2f*>(&lbr[fb0 + 32]);  // subtile 1
      v2f bi1 = *reinterpret_cast<const v2f*>(&lbi[fb0 + 32]);
      v2f bs1 = br1 + bi1;
      rr0 = __builtin_amdgcn_wmma_f32_16x16x4_f32(false, a_r, false, br0,
                                                  (short)0, rr0, false, false);
      ii0 = __builtin_amdgcn_wmma_f32_16x16x4_f32(false, a_i, false, bi0,
                                                  (short)0, ii0, false, false);
      xx0 = __builtin_amdgcn_wmma_f32_16x16x4_f32(false, a_s, false, bs0,
                                                  (short)0, xx0, false, false);
      rr1 = __builtin_amdgcn_wmma_f32_16x16x4_f32(false, a_r, false, br1,
                                                  (short)0, rr1, false, false);
      ii1 = __builtin_amdgcn_wmma_f32_16x16x4_f32(false, a_i, false, bi1,
                                                  (short)0, ii1, false, false);
      xx1 = __builtin_amdgcn_wmma_f32_16x16x4_f32(false, a_s, false, bs1,
                                                  (short)0, xx1, false, false);
    }
    __syncthreads();
  }

  const int crow0 = m0 + (lane >> 4) * 8;
  float* __restrict__ Cr = cr_g + (size_t)img * (size_t)NPIX;
  float* __restrict__ Ci = ci_g + (size_t)img * (size_t)NPIX;
#pragma unroll
  for (int v = 0; v < 8; ++v) {
    size_t o0 = (size_t)(crow0 + v) * DIM + (n0 + bn);
    Cr[o0]      = rr0[v] - ii0[v];
    Ci[o0]      = xx0[v] - rr0[v] - ii0[v];
    Cr[o0 + 16] = rr1[v] - ii1[v];
    Ci[o0 + 16] = xx1[v] - rr1[v] - ii1[v];
  }
}

// ---------------- mean-mode intensity -> masked class scores ----------------
__global__ __launch_bounds__(256) void zero_kernel(float* __restrict__ out, int n) {
  int i = blockIdx.x * blockDim.x + threadIdx.x;
  if (i < n) out[i] = 0.0f;
}

__global__ __launch_bounds__(256) void scores_kernel(
    const float* __restrict__ ur, const float* __restrict__ ui,
    const float* __restrict__ mask, float* __restrict__ out) {
  __shared__ float cls[NCLS];
  const int b   = blockIdx.y;
  const int tid = threadIdx.x;
  if (tid < NCLS) cls[tid] = 0.0f;
  __syncthreads();

  float acc[NCLS];
#pragma unroll
  for (int c = 0; c < NCLS; ++c) acc[c] = 0.0f;

  const float inv_m = 1.0f / (float)NMODE;
  for (int p = blockIdx.x * blockDim.x + tid; p < NPIX;
       p += blockDim.x * gridDim.x) {
    float inten = 0.0f;
#pragma unroll
    for (int m = 0; m < NMODE; ++m) {
      size_t o = (size_t)(m * NBAT + b) * NPIX + p;
      float r = ur[o], im = ui[o];
      inten += r * r + im * im;
    }
    inten *= inv_m;
#pragma unroll
    for (int c = 0; c < NCLS; ++c) acc[c] += inten * mask[(size_t)c * NPIX + p];
  }
#pragma unroll
  for (int c = 0; c < NCLS; ++c) atomicAdd(&cls[c], acc[c]);   // ds_add_f32
  __syncthreads();
  if (tid < NCLS) atomicAdd(&out[b * NCLS + tid], cls[tid]);
}

// ============================================================================
extern "C" void kernel_launch(void* const* d_in, const int* in_sizes, int n_in,
                              void* d_out, int out_size, void* d_ws, size_t ws_size,
                              hipStream_t stream) {
  const float* x_amp     = (const float*)d_in[0];  // [16,1,512,512]
  const float* pnoise    = (const float*)d_in[1];  // [9,16,1,512,512]
  const float* amplitude = (const float*)d_in[2];  // [3,1,1,512,512]
  const float* phase     = (const float*)d_in[3];  // [3,1,1,512,512]
  const float* dmask     = (const float*)d_in[4];  // [10,1,512,512]
  float* out = (float*)d_out;                      // [16,10]
  float* ws  = (float*)d_ws;

  float* wfr = ws;
  float* wfi = wfr + NPIX;
  float* wbr = wfi + NPIX;
  float* wbi = wbr + NPIX;
  float* hr  = wbi + NPIX;
  float* hi  = hr + NPIX;
  const size_t plane = (size_t)NIMG * NPIX;
  float* u_r = hi + NPIX;
  float* u_i = u_r + plane;
  float* v_r = u_i + plane;
  float* v_i = v_r + plane;

  const unsigned ew_blocks = (unsigned)(((size_t)NIMG * NPIX) / 256);

  init_consts_kernel<<<NPIX / 256, 256, 0, stream>>>(wfr, wfi, wbr, wbi, hr, hi);
  init_field_kernel<<<ew_blocks, 256, 0, stream>>>(x_amp, pnoise, u_r, u_i);

  dim3 gblk(32, 8, 1);
  dim3 ggrd(DIM / 32, (DIM / 16) / 8, NIMG);
  const unsigned long long sb = (unsigned long long)NPIX;

  for (int l = 0; l < 4; ++l) {
    if (l < 3)
      ew_layer_kernel<<<ew_blocks, 256, 0, stream>>>(
          u_r, u_i, amplitude + (size_t)l * NPIX, phase + (size_t)l * NPIX);
    // V = Wf * U   (transform rows axis)
    gemm_cplx_kernel<<<ggrd, gblk, 0, stream>>>(wfr, wfi, 0ull, u_r, u_i, sb, v_r, v_i);
    // U = V * Wf   (transform cols axis) -> frequency domain
    gemm_cplx_kernel<<<ggrd, gblk, 0, stream>>>(v_r, v_i, sb, wfr, wfi, 0ull, u_r, u_i);
    ew_h_kernel<<<ew_blocks, 256, 0, stream>>>(u_r, u_i, hr, hi);
    // inverse transform (Wb includes 1/512 each application)
    gemm_cplx_kernel<<<ggrd, gblk, 0, stream>>>(wbr, wbi, 0ull, u_r, u_i, sb, v_r, v_i);
    gemm_cplx_kernel<<<ggrd, gblk, 0, stream>>>(v_r, v_i, sb, wbr, wbi, 0ull, u_r, u_i);
  }

  zero_kernel<<<1, 256, 0, stream>>>(out, NBAT * NCLS);
  scores_kernel<<<dim3(128, NBAT), 256, 0, stream>>>(u_r, u_i, dmask, out);
}